// TwoTower_44298292691577
// MI455X (gfx1250) — compile-verified
//
#include <hip/hip_runtime.h>

// Two-tower gather + row-wise dot:
//   out[b] = dot(W_cust[x[b]], W_art[y[b]]),  D = 64, f32.
//
// Memory-bound (~524 MB @ 23.3 TB/s => ~22.5 us floor). Each wave produces 16
// outputs; rows are read as 128-bit loads (full 256B row per lane pair); the
// cross-lane reduction is done with one V_WMMA_F32_16X16X4_F32 (B = ones):
//   A[m][k] = partial dot of row m over element block k (16 elems each)
//   D[m][n] = sum_k A[m][k] = out[m]  for all n.

typedef float v2f __attribute__((ext_vector_type(2)));
typedef float v8f __attribute__((ext_vector_type(8)));

__device__ __forceinline__ float dot4(float4 a, float4 b) {
    return fmaf(a.x, b.x, fmaf(a.y, b.y, fmaf(a.z, b.z, a.w * b.w)));
}

__global__ __launch_bounds__(256) void twotower_dot_wmma(
    const int* __restrict__ x, const int* __restrict__ y,
    const float* __restrict__ Wc, const float* __restrict__ Wa,
    float* __restrict__ out, int Bn)
{
    const int lane   = threadIdx.x & 31;
    const int waveIn = threadIdx.x >> 5;          // 8 waves / block (wave32)
    const long long waveId = (long long)blockIdx.x * 8 + waveIn;
    const long long b0 = waveId * 16;             // 16 outputs per wave
    if (b0 >= Bn) return;

    // Row assignment: lanes m and m+16 both handle output b0+m.
    const int m = lane & 15;
    long long b = b0 + m;
    if (b >= Bn) b = Bn - 1;                      // safe clamp (B % 16 == 0 normally)

    const int xi = x[b];
    const int yi = y[b];
    const float4* __restrict__ cr = (const float4*)(Wc + (size_t)xi * 64);
    const float4* __restrict__ ar = (const float4*)(Wa + (size_t)yi * 64);

    // lanes 0-15: float4 chunks 0..7  (elements  0..31 -> K=0, K=1)
    // lanes16-31: float4 chunks 8..15 (elements 32..63 -> K=2, K=3)
    const int c0 = (lane >> 4) * 8;

    float4 c4[8], a4[8];
#pragma unroll
    for (int i = 0; i < 8; ++i) { c4[i] = cr[c0 + i]; a4[i] = ar[c0 + i]; }

    float p0 = 0.f, p1 = 0.f;
#pragma unroll
    for (int i = 0; i < 4; ++i) p0 += dot4(c4[i], a4[i]);
#pragma unroll
    for (int i = 0; i < 4; ++i) p1 += dot4(c4[4 + i], a4[4 + i]);

    // A-matrix (16x4 f32): VGPR0 = K{0|2}, VGPR1 = K{1|3} per ISA layout.
    v2f A;  A.x = p0;  A.y = p1;
    v2f Bm; Bm.x = 1.0f; Bm.y = 1.0f;             // B = ones (4x16), layout-invariant
    v8f C = {};
    C = __builtin_amdgcn_wmma_f32_16x16x4_f32(
        /*neg_a=*/false, A, /*neg_b=*/false, Bm,
        /*c_mod=*/(short)0, C, /*reuse_a=*/false, /*reuse_b=*/false);

    // D layout: lane<16 -> C[j] = out[j]; lane>=16 -> C[j] = out[j+8].
    const int sel = lane & 7;
    float v = C[0];
    v = (sel == 1) ? C[1] : v;
    v = (sel == 2) ? C[2] : v;
    v = (sel == 3) ? C[3] : v;
    v = (sel == 4) ? C[4] : v;
    v = (sel == 5) ? C[5] : v;
    v = (sel == 6) ? C[6] : v;
    v = (sel == 7) ? C[7] : v;

    if (lane < 8) {
        long long o = b0 + sel;
        if (o < Bn) out[o] = v;
    } else if (lane >= 16 && lane < 24) {
        long long o = b0 + 8 + sel;
        if (o < Bn) out[o] = v;
    }
}

extern "C" void kernel_launch(void* const* d_in, const int* in_sizes, int n_in,
                              void* d_out, int out_size, void* d_ws, size_t ws_size,
                              hipStream_t stream) {
    const int*   x  = (const int*)d_in[0];
    const int*   y  = (const int*)d_in[1];
    const float* Wc = (const float*)d_in[2];
    const float* Wa = (const float*)d_in[3];
    float*       out = (float*)d_out;

    const int Bn = in_sizes[0];
    const long long waves  = ((long long)Bn + 15) / 16;
    const int       blocks = (int)((waves + 7) / 8);

    twotower_dot_wmma<<<blocks, 256, 0, stream>>>(x, y, Wc, Wa, out, Bn);
}